// Attention_84499186582044
// MI455X (gfx1250) — compile-verified
//
#include <hip/hip_runtime.h>
#include <hip/hip_bf16.h>
#include <math.h>

// ---------------- problem constants (from reference) ----------------
#define B_      32
#define S_      1
#define H_      32
#define KVH_    8
#define D_      128
#define HID_    4096
#define MAXSEQ_ 2048
#define CURPOS_ (MAXSEQ_ - 1)
#define GROUPS_ (H_ / KVH_)      // 4
#define EPS_    1e-6f
#define NCHUNK_ 16               // flash-decode split of the 2048 KV positions
#define PSTRIDE_ (8 + GROUPS_ * D_)  // per-(b,kvh,chunk) partial: m[4], l[4], acc[4][128]

typedef __attribute__((ext_vector_type(2))) float v2f;
typedef __attribute__((ext_vector_type(8))) float v8f;

// =====================================================================
// WMMA f32 GEMM:  C[M=32, N] = X[32, K] * W[N, K]^T   (all row-major)
// One wave owns one 16x16 C tile; inner dim walked 4 at a time via
// V_WMMA_F32_16X16X4_F32. Per-lane float2 loads match the ISA layout:
//   A 16x4 : lane(0..15)=row M, VGPR v = K(v);  lanes 16..31 = K(v+2)
//   B 4x16 : lane(0..15)=col N, VGPR v = K(v);  lanes 16..31 = K(v+2)
//   C/D    : VGPR j, lanes0-15 -> M=j, lanes16-31 -> M=j+8
// Memory-bound kernel: W is streamed exactly once (the dominant bytes);
// prefetch runs ~4 KB ahead of the weight stream.
// =====================================================================
__global__ __launch_bounds__(256) void gemm_xwt_wmma(
    const float* __restrict__ X, const float* __restrict__ W,
    float* __restrict__ C, int N, int K)
{
  const int wave = threadIdx.x >> 5;      // 8 waves / block
  const int lane = threadIdx.x & 31;
  const int lo   = lane & 15;
  const int hi   = lane >> 4;             // 0 or 1
  const int n0   = (blockIdx.x * 8 + wave) * 16;
  const int m0   = blockIdx.y * 16;
  if (n0 >= N) return;                    // wave-uniform: EXEC stays all-ones

  const float* xrow = X + (size_t)(m0 + lo) * K + 2 * hi;
  const float* wrow = W + (size_t)(n0 + lo) * K + 2 * hi;

  v8f c = {0.f, 0.f, 0.f, 0.f, 0.f, 0.f, 0.f, 0.f};
  for (int k0 = 0; k0 < K; k0 += 16) {
    __builtin_prefetch(wrow + k0 + 1024, 0, 0);   // global_prefetch_b8, ~4KB ahead
#pragma unroll
    for (int u = 0; u < 4; ++u) {
      v2f a = *(const v2f*)(xrow + k0 + 4 * u);
      v2f b = *(const v2f*)(wrow + k0 + 4 * u);
      c = __builtin_amdgcn_wmma_f32_16x16x4_f32(
              /*neg_a=*/false, a, /*neg_b=*/false, b,
              /*c_mod=*/(short)0, c, /*reuse_a=*/false, /*reuse_b=*/false);
    }
  }

  float* crow = C + (size_t)(m0 + 8 * hi) * N + n0 + lo;
#pragma unroll
  for (int j = 0; j < 8; ++j) crow[(size_t)j * N] = c[j];
}

// =====================================================================
// Per-head RMSNorm + RoPE (in place). One 128-thread block per head vec.
// =====================================================================
__global__ __launch_bounds__(128) void norm_rope_kernel(
    float* __restrict__ qbuf,      // [B, H*D]
    float* __restrict__ kbuf,      // [B, KVH*D]
    const float* __restrict__ qw, const float* __restrict__ kw,
    const float* __restrict__ cosT, const float* __restrict__ sinT,
    const int* __restrict__ pid)
{
  __shared__ float red[D_];
  __shared__ float xs[D_];
  const int b  = blockIdx.x / (H_ + KVH_);
  const int hh = blockIdx.x % (H_ + KVH_);
  float* vec; const float* w;
  if (hh < H_) { vec = qbuf + ((size_t)b * H_ + hh) * D_;          w = qw; }
  else         { vec = kbuf + ((size_t)b * KVH_ + (hh - H_)) * D_; w = kw; }

  const int d = threadIdx.x;
  float xv = vec[d];
  red[d] = xv * xv;
  __syncthreads();
  for (int s2 = 64; s2 > 0; s2 >>= 1) {
    if (d < s2) red[d] += red[d + s2];
    __syncthreads();
  }
  const float rms = rsqrtf(red[0] * (1.0f / D_) + EPS_);
  const float xn  = xv * rms * w[d];
  xs[d] = xn;
  __syncthreads();

  const int p = pid[b * S_];                       // CUR_POS for every batch
  const float c = cosT[(size_t)p * D_ + d];
  const float s = sinT[(size_t)p * D_ + d];
  const float rot = (d < D_ / 2) ? -xs[d + D_ / 2] : xs[d - D_ / 2];
  vec[d] = xn * c + rot * s;
}

// =====================================================================
// Flash-decode attention, stage 1.
// grid = (B*KVH, NCHUNK), block = 256 (8 waves).
// Each wave streams 16 KV rows; lane owns d-slice [4*lane, 4*lane+4):
// coalesced float4 loads of K and V (the 512 MB stream that dominates).
// Scores reduced with __shfl_xor butterflies; online softmax per wave;
// waves merged in LDS; (m, l, acc) partial written per (b,kvh,chunk).
// Position CUR_POS uses the freshly projected k/v (cache not mutated).
// =====================================================================
__global__ __launch_bounds__(256) void attn_partial_kernel(
    const float* __restrict__ qbuf,     // [B, H*D]   (normed + roped)
    const float* __restrict__ knew,     // [B, KVH*D] (normed + roped)
    const float* __restrict__ vnew,     // [B, KVH*D]
    const float* __restrict__ kcache,   // [1,B,MAXSEQ,KVH,D]
    const float* __restrict__ vcache,
    float* __restrict__ partials)       // [B*KVH, NCHUNK, PSTRIDE]
{
  __shared__ float sm[8][GROUPS_];
  __shared__ float sl[8][GROUPS_];
  __shared__ float sacc[8][GROUPS_][D_];

  const int b    = blockIdx.x / KVH_;
  const int kvh  = blockIdx.x % KVH_;
  const int wave = threadIdx.x >> 5;
  const int lane = threadIdx.x & 31;
  const float scale = 0.088388347648318447f;   // 1/sqrt(128)

  // q for the 4 grouped heads, this lane's d-slice
  float4 q4[GROUPS_];
#pragma unroll
  for (int g = 0; g < GROUPS_; ++g) {
    const float* qh = qbuf + ((size_t)b * H_ + kvh * GROUPS_ + g) * D_;
    q4[g] = *(const float4*)(qh + 4 * lane);
  }

  float m[GROUPS_], l[GROUPS_], acc[GROUPS_][4];
#pragma unroll
  for (int g = 0; g < GROUPS_; ++g) {
    m[g] = -1e30f; l[g] = 0.f;
    acc[g][0] = acc[g][1] = acc[g][2] = acc[g][3] = 0.f;
  }

  const int tpw = MAXSEQ_ / (NCHUNK_ * 8);     // 16 positions per wave
  const int t0  = blockIdx.y * (MAXSEQ_ / NCHUNK_) + wave * tpw;

  for (int i = 0; i < tpw; ++i) {
    const int t = t0 + i;
    const size_t coff = (((size_t)b * MAXSEQ_ + t) * KVH_ + kvh) * D_;
    const size_t noff = ((size_t)b * KVH_ + kvh) * D_;
    const float* kr = (t == CURPOS_) ? (knew + noff) : (kcache + coff);
    const float* vr = (t == CURPOS_) ? (vnew + noff) : (vcache + coff);

    const float4 kk = *(const float4*)(kr + 4 * lane);
    float s[GROUPS_];
#pragma unroll
    for (int g = 0; g < GROUPS_; ++g)
      s[g] = q4[g].x * kk.x + q4[g].y * kk.y + q4[g].z * kk.z + q4[g].w * kk.w;
#pragma unroll
    for (int off = 16; off > 0; off >>= 1) {
#pragma unroll
      for (int g = 0; g < GROUPS_; ++g) s[g] += __shfl_xor(s[g], off, 32);
    }

    const float4 vv = *(const float4*)(vr + 4 * lane);
#pragma unroll
    for (int g = 0; g < GROUPS_; ++g) {
      const float sc = s[g] * scale;
      const float nm = fmaxf(m[g], sc);
      const float f  = __expf(m[g] - nm);
      const float p  = __expf(sc - nm);
      l[g] = l[g] * f + p;
      acc[g][0] = acc[g][0] * f + p * vv.x;
      acc[g][1] = acc[g][1] * f + p * vv.y;
      acc[g][2] = acc[g][2] * f + p * vv.z;
      acc[g][3] = acc[g][3] * f + p * vv.w;
      m[g] = nm;
    }
  }

  // dump per-wave state to LDS
#pragma unroll
  for (int g = 0; g < GROUPS_; ++g) {
    sacc[wave][g][4 * lane + 0] = acc[g][0];
    sacc[wave][g][4 * lane + 1] = acc[g][1];
    sacc[wave][g][4 * lane + 2] = acc[g][2];
    sacc[wave][g][4 * lane + 3] = acc[g][3];
    if (lane == 0) { sm[wave][g] = m[g]; sl[wave][g] = l[g]; }
  }
  __syncthreads();

  // merge the 8 waves, write chunk partial
  float* base = partials +
      ((size_t)blockIdx.x * NCHUNK_ + blockIdx.y) * PSTRIDE_;
  for (int it = threadIdx.x; it < GROUPS_ * D_; it += blockDim.x) {
    const int g = it / D_, d = it % D_;
    float M = -1e30f;
#pragma unroll
    for (int w = 0; w < 8; ++w) M = fmaxf(M, sm[w][g]);
    float L = 0.f, A = 0.f;
#pragma unroll
    for (int w = 0; w < 8; ++w) {
      const float f = __expf(sm[w][g] - M);
      L += sl[w][g] * f;
      A += sacc[w][g][d] * f;
    }
    base[8 + g * D_ + d] = A;
    if (d == 0) { base[g] = M; base[GROUPS_ + g] = L; }
  }
}

// =====================================================================
// Flash-decode attention, stage 2: merge the NCHUNK partials.
// grid = B*H, block = 128 (one thread per d).
// =====================================================================
__global__ __launch_bounds__(128) void attn_combine_kernel(
    const float* __restrict__ partials, float* __restrict__ attn_out)
{
  const int b   = blockIdx.x / H_;
  const int h   = blockIdx.x % H_;
  const int kvh = h / GROUPS_;
  const int g   = h % GROUPS_;
  const float* pb = partials + (size_t)(b * KVH_ + kvh) * NCHUNK_ * PSTRIDE_;

  float M = -1e30f;
#pragma unroll
  for (int c = 0; c < NCHUNK_; ++c) M = fmaxf(M, pb[c * PSTRIDE_ + g]);
  float L = 0.f;
#pragma unroll
  for (int c = 0; c < NCHUNK_; ++c)
    L += pb[c * PSTRIDE_ + GROUPS_ + g] * __expf(pb[c * PSTRIDE_ + g] - M);

  const int d = threadIdx.x;
  float A = 0.f;
#pragma unroll
  for (int c = 0; c < NCHUNK_; ++c)
    A += pb[c * PSTRIDE_ + 8 + g * D_ + d] * __expf(pb[c * PSTRIDE_ + g] - M);

  attn_out[(size_t)b * (H_ * D_) + h * D_ + d] = A / L;
}

// =====================================================================
extern "C" void kernel_launch(void* const* d_in, const int* in_sizes, int n_in,
                              void* d_out, int out_size, void* d_ws, size_t ws_size,
                              hipStream_t stream) {
  const float* x    = (const float*)d_in[0];
  const float* wq   = (const float*)d_in[1];
  const float* wk   = (const float*)d_in[2];
  const float* wv   = (const float*)d_in[3];
  const float* wo   = (const float*)d_in[4];
  const float* qnw  = (const float*)d_in[5];
  const float* knw  = (const float*)d_in[6];
  const float* cosT = (const float*)d_in[7];
  const float* sinT = (const float*)d_in[8];
  const float* kc   = (const float*)d_in[9];
  const float* vc   = (const float*)d_in[10];
  const int*   pid  = (const int*)d_in[11];
  float* out = (float*)d_out;

  // workspace carve-up (floats): ~9.4 MB total
  float* ws       = (float*)d_ws;
  float* qbuf     = ws;                                   // 32*4096
  float* kbuf     = qbuf + (size_t)B_ * H_ * D_;          // 32*1024
  float* vbuf     = kbuf + (size_t)B_ * KVH_ * D_;        // 32*1024
  float* attn_out = vbuf + (size_t)B_ * KVH_ * D_;        // 32*4096
  float* partials = attn_out + (size_t)B_ * H_ * D_;      // 256*16*520

  // 1) QKV projections (WMMA f32 GEMMs)
  gemm_xwt_wmma<<<dim3(HID_ / 128, 2), 256, 0, stream>>>(x, wq, qbuf, H_ * D_, HID_);
  gemm_xwt_wmma<<<dim3((KVH_ * D_) / 128, 2), 256, 0, stream>>>(x, wk, kbuf, KVH_ * D_, HID_);
  gemm_xwt_wmma<<<dim3((KVH_ * D_) / 128, 2), 256, 0, stream>>>(x, wv, vbuf, KVH_ * D_, HID_);

  // 2) RMSNorm + RoPE on q and k heads
  norm_rope_kernel<<<B_ * (H_ + KVH_), 128, 0, stream>>>(
      qbuf, kbuf, qnw, knw, cosT, sinT, pid);

  // 3) flash-decode partials over the KV cache
  attn_partial_kernel<<<dim3(B_ * KVH_, NCHUNK_), 256, 0, stream>>>(
      qbuf, kbuf, vbuf, kc, vc, partials);

  // 4) merge chunks -> attention output [B, H*D]
  attn_combine_kernel<<<B_ * H_, 128, 0, stream>>>(partials, attn_out);

  // 5) output projection (WMMA f32 GEMM) -> d_out
  gemm_xwt_wmma<<<dim3(HID_ / 128, 2), 256, 0, stream>>>(attn_out, wo, out, HID_, HID_);
}